// AttnBlockpp1d_70463233458257
// MI455X (gfx1250) — compile-verified
//
#include <hip/hip_runtime.h>
#include <hip/hip_bf16.h>

#define BATCH 8
#define CDIM 512
#define TDIM 1024
#define NHEADS 8
#define CH 64
#define NGROUPS 32
#define CPG (CDIM / NGROUPS)   // 16 channels per group
#define EPSV 1e-6f

typedef __attribute__((ext_vector_type(2))) float v2f;
typedef __attribute__((ext_vector_type(8))) float v8f;

__device__ __forceinline__ v8f v8zero() {
    v8f z;
#pragma unroll
    for (int i = 0; i < 8; ++i) z[i] = 0.0f;
    return z;
}

// D = A(16x4) * B(4x16) + C, fp32 WMMA (CDNA5)
__device__ __forceinline__ v8f wmma4(v2f a, v2f b, v8f c) {
    return __builtin_amdgcn_wmma_f32_16x16x4_f32(
        /*neg_a=*/false, a, /*neg_b=*/false, b,
        /*c_mod=*/(short)0, c, /*reuse_a=*/false, /*reuse_b=*/false);
}

// aligned v2f load from LDS (row stride and kb chosen so addr is 8B aligned)
__device__ __forceinline__ v2f lds2(const float* p) {
    return *(const v2f*)p;
}

// ---------------------------------------------------------------------------
// Kernel 1: GroupNorm.  One block per (batch, group).  CPG*T = 16384 elems.
// ---------------------------------------------------------------------------
__global__ __launch_bounds__(256) void gn_kernel(
    const float* __restrict__ x, const float* __restrict__ gamma,
    const float* __restrict__ beta, float* __restrict__ h) {
    const int b = blockIdx.x / NGROUPS;
    const int g = blockIdx.x % NGROUPS;
    const int tid = threadIdx.x;
    const size_t base = ((size_t)b * CDIM + (size_t)g * CPG) * TDIM;
    const float* xp = x + base;

    float s = 0.f, ss = 0.f;
    for (int i = tid; i < CPG * TDIM; i += 256) {
        float v = xp[i];
        s += v;
        ss += v * v;
    }
    __shared__ float rs[256], rss[256];
    rs[tid] = s;
    rss[tid] = ss;
    __syncthreads();
    for (int off = 128; off > 0; off >>= 1) {
        if (tid < off) {
            rs[tid] += rs[tid + off];
            rss[tid] += rss[tid + off];
        }
        __syncthreads();
    }
    const float inv_n = 1.0f / (float)(CPG * TDIM);
    const float mu = rs[0] * inv_n;
    const float var = rss[0] * inv_n - mu * mu;
    const float rstd = rsqrtf(var + EPSV);

    float* hp = h + base;
    for (int i = tid; i < CPG * TDIM; i += 256) {
        int c = g * CPG + (i / TDIM);
        hp[i] = (xp[i] - mu) * rstd * gamma[c] + beta[c];
    }
}

// ---------------------------------------------------------------------------
// Kernel 2: NIN projection GEMM.  out[b,d,t] = sum_c W[c,d]*in[b,c,t] + bias[d]
//           (+ res[b,d,t] if res != nullptr).
// grid = (T/128, C/128, B), block = 256 (8 waves).  Block: 128(d) x 128(t).
// Wave w: d-tiles {2*(w&3), 2*(w&3)+1}, t-tiles {4*(w>>2) .. +3}; 8 accums.
// LDS tiles stored K-minor (stride 20) so fragments are single b64 loads.
// ---------------------------------------------------------------------------
#define PK_TK 16
#define PK_LD 20   // padded row stride (floats): 8B aligned, conflict-free
__global__ __launch_bounds__(256) void proj_kernel(
    const float* __restrict__ in, const float* __restrict__ W,
    const float* __restrict__ bias, const float* __restrict__ res,
    float* __restrict__ out) {
    __shared__ float Wtr[128][PK_LD];   // [d][k]
    __shared__ float Htr[128][PK_LD];   // [t][k]

    const int b = blockIdx.z;
    const int d0 = blockIdx.y * 128;
    const int t0 = blockIdx.x * 128;
    const int tid = threadIdx.x;
    const int wave = tid >> 5;
    const int lane = tid & 31;
    const int dgrp = wave & 3;    // owns d-tiles 2*dgrp, 2*dgrp+1
    const int tgrp = wave >> 2;   // owns t-tiles 4*tgrp .. 4*tgrp+3
    const int lm = lane & 15;
    const int khalf = (lane >> 4) * 2;

    const float* inb = in + (size_t)b * CDIM * TDIM;
    v8f acc[2][4];
#pragma unroll
    for (int di = 0; di < 2; ++di)
#pragma unroll
        for (int j = 0; j < 4; ++j) acc[di][j] = v8zero();

    for (int k0 = 0; k0 < CDIM; k0 += PK_TK) {
        // stage W tile (16k x 128d) and h tile (16k x 128t), K-minor in LDS
        for (int i = tid; i < PK_TK * 128; i += 256) {
            const int kk = i >> 7, dd = i & 127;
            Wtr[dd][kk] = W[(size_t)(k0 + kk) * CDIM + d0 + dd];
        }
        for (int i = tid; i < PK_TK * 128; i += 256) {
            const int kk = i >> 7, tt = i & 127;
            Htr[tt][kk] = inb[(size_t)(k0 + kk) * TDIM + t0 + tt];
        }
        if (k0 + PK_TK < CDIM) {
            __builtin_prefetch(inb + (size_t)(k0 + PK_TK + (tid >> 4)) * TDIM + t0, 0, 1);
        }
        __syncthreads();

#pragma unroll
        for (int kk = 0; kk < PK_TK; kk += 4) {
            const int kb = kk + khalf;
            v2f a[2], bb[4];
#pragma unroll
            for (int di = 0; di < 2; ++di)
                a[di] = lds2(&Wtr[16 * (2 * dgrp + di) + lm][kb]);
#pragma unroll
            for (int j = 0; j < 4; ++j)
                bb[j] = lds2(&Htr[16 * (4 * tgrp + j) + lm][kb]);
#pragma unroll
            for (int di = 0; di < 2; ++di)
#pragma unroll
                for (int j = 0; j < 4; ++j)
                    acc[di][j] = wmma4(a[di], bb[j], acc[di][j]);
        }
        __syncthreads();
    }

    // epilogue: C/D layout -> row M = r + 8*(lane>>4), col N = lane&15
    float* outb = out + (size_t)b * CDIM * TDIM;
    const int rowhalf = (lane >> 4) * 8;
#pragma unroll
    for (int di = 0; di < 2; ++di) {
#pragma unroll
        for (int j = 0; j < 4; ++j) {
            const int t = t0 + 16 * (4 * tgrp + j) + lm;
#pragma unroll
            for (int r = 0; r < 8; ++r) {
                const int d = d0 + 16 * (2 * dgrp + di) + rowhalf + r;
                float v = acc[di][j][r] + bias[d];
                if (res) v += res[(size_t)b * CDIM * TDIM + (size_t)d * TDIM + t];
                outb[(size_t)d * TDIM + t] = v;
            }
        }
    }
}

// ---------------------------------------------------------------------------
// Kernel 3: flash attention per (b, head, 64-query block).
// grid = (T/64, B*H), block = 256 (8 waves).
// S = (Q^T K) * Ch^-0.5 via WMMA; online softmax; A += V * P^T via WMMA.
// Ktr stored [i][c] (K-minor for B frags); Vt [c][i]; Pb [t][i].
// ---------------------------------------------------------------------------
#define AT_LD 68   // padded row stride (floats): 8B aligned, conflict-free
__global__ __launch_bounds__(256) void attn_kernel(
    const float* __restrict__ q, const float* __restrict__ k,
    const float* __restrict__ v, float* __restrict__ a_out) {
    __shared__ float Ktr[64][AT_LD];  // K^T: [i][c]
    __shared__ float Vt[64][AT_LD];   // V: [c][i]
    __shared__ float Pb[64][AT_LD];   // S then P: [t][i]
    __shared__ float mrow[64], lrow[64], crow[64];

    const int t0 = blockIdx.x * 64;
    const int bh = blockIdx.y;
    const int b = bh / NHEADS;
    const int hd = bh % NHEADS;
    const size_t head_base = ((size_t)b * CDIM + (size_t)hd * CH) * TDIM;
    const float* qp = q + head_base;
    const float* kp = k + head_base;
    const float* vp = v + head_base;

    const int tid = threadIdx.x;
    const int wave = tid >> 5;
    const int lane = tid & 31;
    const int wr = wave & 3;             // row tile (t for S, c for output)
    const int wcbase = (wave >> 2) * 2;  // first of two column tiles
    const int lm = lane & 15;
    const int khalf = (lane >> 4) * 2;
    const int rowhalf = (lane >> 4) * 8;

    if (tid < 64) {
        mrow[tid] = -1e30f;
        lrow[tid] = 0.0f;
    }

    // Preload Q fragments for this wave's S row tile: A[m=t][k=c] = Q[c][t]
    v2f qf[16];
    {
        const int tm = t0 + 16 * wr + lm;
#pragma unroll
        for (int kk = 0; kk < 16; ++kk) {
            const int cc = 4 * kk + khalf;
            qf[kk].x = qp[(size_t)cc * TDIM + tm];
            qf[kk].y = qp[(size_t)(cc + 1) * TDIM + tm];
        }
    }

    v8f oacc[2];
    oacc[0] = v8zero();
    oacc[1] = v8zero();
    __syncthreads();

    for (int ib = 0; ib < TDIM / 64; ++ib) {
        const int i0 = ib * 64;
        // stage K (transposed) and V tiles; global reads coalesced along i
        for (int idx = tid; idx < 64 * 64; idx += 256) {
            const int c = idx >> 6, i = idx & 63;
            const float kvv = kp[(size_t)c * TDIM + i0 + i];
            Ktr[i][c] = kvv;
            Vt[c][i] = vp[(size_t)c * TDIM + i0 + i];
        }
        __syncthreads();

        // ---- S = Q^T K (this wave: row tile wr, col tiles wcbase, wcbase+1)
#pragma unroll
        for (int j = 0; j < 2; ++j) {
            const int tc = wcbase + j;
            v8f sacc = v8zero();
            const int ni = 16 * tc + lm;
#pragma unroll
            for (int kk = 0; kk < 16; ++kk) {
                const v2f bb = lds2(&Ktr[ni][4 * kk + khalf]);
                sacc = wmma4(qf[kk], bb, sacc);
            }
            // scatter to LDS with scale Ch^-0.5 = 0.125
#pragma unroll
            for (int r = 0; r < 8; ++r) {
                const int tt = 16 * wr + rowhalf + r;
                Pb[tt][ni] = sacc[r] * 0.125f;
            }
        }
        __syncthreads();

        // ---- online softmax: thread t (< 64) owns query row t
        if (tid < 64) {
            float mb = -1e30f;
#pragma unroll 4
            for (int i = 0; i < 64; ++i) mb = fmaxf(mb, Pb[tid][i]);
            const float mnew = fmaxf(mrow[tid], mb);
            const float corr = __expf(mrow[tid] - mnew);
            float sum = 0.f;
#pragma unroll 4
            for (int i = 0; i < 64; ++i) {
                const float p = __expf(Pb[tid][i] - mnew);
                Pb[tid][i] = p;
                sum += p;
            }
            lrow[tid] = lrow[tid] * corr + sum;
            mrow[tid] = mnew;
            crow[tid] = corr;
        }
        __syncthreads();

        // ---- acc rescale + A += V * P^T (c tile wr, t tiles wcbase..+1)
#pragma unroll
        for (int j = 0; j < 2; ++j) {
            const int tc = wcbase + j;
            const int nt = 16 * tc + lm;     // t column
            const float corr = crow[nt];
#pragma unroll
            for (int r = 0; r < 8; ++r) oacc[j][r] *= corr;
            const int mc = 16 * wr + lm;     // c row
#pragma unroll
            for (int kk = 0; kk < 16; ++kk) {
                const int kb = 4 * kk + khalf;
                const v2f aa = lds2(&Vt[mc][kb]);
                const v2f bb = lds2(&Pb[nt][kb]);
                oacc[j] = wmma4(aa, bb, oacc[j]);
            }
        }
        __syncthreads();
    }

    // finalize: divide by l, write a[b, h*CH + c, t0 + t]
    float* ap = a_out + head_base;
#pragma unroll
    for (int j = 0; j < 2; ++j) {
        const int tc = wcbase + j;
        const int tl = 16 * tc + lm;
        const float linv = 1.0f / lrow[tl];
#pragma unroll
        for (int r = 0; r < 8; ++r) {
            const int c = 16 * wr + rowhalf + r;
            ap[(size_t)c * TDIM + t0 + tl] = oacc[j][r] * linv;
        }
    }
}

// ---------------------------------------------------------------------------
extern "C" void kernel_launch(void* const* d_in, const int* in_sizes, int n_in,
                              void* d_out, int out_size, void* d_ws, size_t ws_size,
                              hipStream_t stream) {
    const float* x     = (const float*)d_in[0];
    const float* gamma = (const float*)d_in[1];
    const float* beta  = (const float*)d_in[2];
    const float* W0 = (const float*)d_in[3];
    const float* b0 = (const float*)d_in[4];
    const float* W1 = (const float*)d_in[5];
    const float* b1 = (const float*)d_in[6];
    const float* W2 = (const float*)d_in[7];
    const float* b2 = (const float*)d_in[8];
    const float* W3 = (const float*)d_in[9];
    const float* b3 = (const float*)d_in[10];
    float* out = (float*)d_out;

    float* wsf = (float*)d_ws;
    const size_t S1 = (size_t)BATCH * CDIM * TDIM;  // 4 Mi floats
    float* hbuf = wsf;
    float* qbuf = wsf + S1;
    float* kbuf = wsf + 2 * S1;
    float* vbuf = wsf + 3 * S1;
    float* abuf = hbuf;  // h no longer needed once attention runs

    gn_kernel<<<BATCH * NGROUPS, 256, 0, stream>>>(x, gamma, beta, hbuf);

    dim3 pg(TDIM / 128, CDIM / 128, BATCH);
    proj_kernel<<<pg, 256, 0, stream>>>(hbuf, W0, b0, nullptr, qbuf);
    proj_kernel<<<pg, 256, 0, stream>>>(hbuf, W1, b1, nullptr, kbuf);
    proj_kernel<<<pg, 256, 0, stream>>>(hbuf, W2, b2, nullptr, vbuf);

    attn_kernel<<<dim3(TDIM / 64, BATCH * NHEADS), 256, 0, stream>>>(qbuf, kbuf, vbuf, abuf);

    proj_kernel<<<pg, 256, 0, stream>>>(abuf, W3, b3, x, out);
}